// PivotEntityPooler_8727373546096
// MI455X (gfx1250) — compile-verified
//
#include <hip/hip_runtime.h>
#include <hip/hip_bf16.h>
#include <stddef.h>

// PivotEntityPooler: out[b,h] = mean_{s in [start_b,end_b)} X[b,s,h]
// B=256, S=512, H=768, fp32. Bandwidth-bound (~100MB read at 23.3 TB/s -> ~4.4us).
// WMMA f32 16x16x4 performs the masked accumulation (A = mask replicated over M,
// B = X tile). s-loop split: masked edge chunks + mask-free interior whose hot
// loop is one s_clause of 8 coalesced b32 loads + 4 WMMAs + 1 pointer add.

#define BB 256
#define SS 512
#define HH 768

typedef __attribute__((ext_vector_type(2))) float v2f;
typedef __attribute__((ext_vector_type(8))) float v8f;

static __device__ __forceinline__ v8f wmma_acc(v2f a, v2f b, v8f c) {
    return __builtin_amdgcn_wmma_f32_16x16x4_f32(
        false, a, false, b, (short)0, c, false, false);
}

__global__ __launch_bounds__(256) void pivot_pool_wmma_kernel(
    const float* __restrict__ X,      // (B,S,H) fp32
    const int*   __restrict__ piv,    // (B,2) int32 [start,end)
    float*       __restrict__ out)    // (B,H) fp32
{
    const int lane = threadIdx.x & 31;
    const int wave = threadIdx.x >> 5;

    // 48 H-tiles of 16; 8 waves per block -> 6 blocks per batch
    const int b    = blockIdx.x / 6;
    const int hgrp = blockIdx.x % 6;
    const int h0   = (hgrp * 8 + wave) * 16;

    // Wave-uniform pivot window (scalarized by the compiler)
    const int start = piv[2 * b];
    const int end   = piv[2 * b + 1];
    const float inv = 1.0f / (float)(end - start);

    // Lane-half selects K pair {0,1} vs {2,3}; VGPR index selects within pair
    // (matches both the f32 A 16x4 and B 4x16 wave32 layouts).
    const int kb  = (lane < 16) ? 0 : 2;
    const int col = lane & 15;
    const float* Xb = X + ((size_t)b * SS * HH) + (size_t)h0 + col;

    v8f acc0 = {}, acc1 = {}, acc2 = {}, acc3 = {};
    v2f ones; ones.x = 1.0f; ones.y = 1.0f;

    const int s0 = start & ~3;   // aligned window begin (<= 252)
    const int e0 = end & ~3;     // aligned window end   (<= 508)

    // ---- prologue: masked chunk [s0, s0+4) — start edge, and the whole
    // window when end <= s0+4. Rows <= 255 < SS. ----
    {
        const int r0 = s0 + kb, r1 = r0 + 1;
        v2f a, bv;
        a.x  = (r0 >= start && r0 < end) ? 1.0f : 0.0f;
        a.y  = (r1 >= start && r1 < end) ? 1.0f : 0.0f;
        bv.x = Xb[(size_t)r0 * HH];
        bv.y = Xb[(size_t)r1 * HH];
        acc0 = wmma_acc(a, bv, acc0);
    }

    // ---- interior: rows [s0+4, e0) fully inside [start,end): mask == 1.
    // Advancing base pointer + immediate offsets (0..13*HH*4 < 2^24 bytes). ----
    const float* p = Xb + (size_t)(s0 + 4 + kb) * HH;
    int s = s0 + 4;
    for (; s + 16 <= e0; s += 16, p += (size_t)16 * HH) {
        v2f b0, b1, b2, b3;
        b0.x = p[0];             b0.y = p[HH];
        b1.x = p[4 * HH];        b1.y = p[5 * HH];
        b2.x = p[8 * HH];        b2.y = p[9 * HH];
        b3.x = p[12 * HH];       b3.y = p[13 * HH];
        acc0 = wmma_acc(ones, b0, acc0);
        acc1 = wmma_acc(ones, b1, acc1);
        acc2 = wmma_acc(ones, b2, acc2);
        acc3 = wmma_acc(ones, b3, acc3);
    }
    for (; s + 4 <= e0; s += 4, p += (size_t)4 * HH) {   // 0..3 residual chunks
        v2f bv;
        bv.x = p[0];
        bv.y = p[HH];
        acc1 = wmma_acc(ones, bv, acc1);
    }

    // ---- epilogue: masked chunk [e0, e0+4) — end edge; skipped when it would
    // overlap the prologue chunk. r >= e0 >= start always, so only r < end.
    // Rows <= 511 < SS. ----
    if (end > e0 && e0 >= s0 + 4) {
        const int r0 = e0 + kb, r1 = r0 + 1;
        v2f a, bv;
        a.x  = (r0 < end) ? 1.0f : 0.0f;
        a.y  = (r1 < end) ? 1.0f : 0.0f;
        bv.x = Xb[(size_t)r0 * HH];
        bv.y = Xb[(size_t)r1 * HH];
        acc2 = wmma_acc(a, bv, acc2);
    }

    // D[m,n] identical over m; D VGPR0 on lanes 0-15 = D[0, n=lane].
    if (lane < 16) {
        out[(size_t)b * HH + h0 + col] =
            (acc0[0] + acc1[0] + acc2[0] + acc3[0]) * inv;
    }
}

extern "C" void kernel_launch(void* const* d_in, const int* in_sizes, int n_in,
                              void* d_out, int out_size, void* d_ws, size_t ws_size,
                              hipStream_t stream) {
    (void)in_sizes; (void)n_in; (void)out_size; (void)d_ws; (void)ws_size;
    const float* X   = (const float*)d_in[0];   // hidden_states (B,S,H) fp32
    const int*   piv = (const int*)d_in[1];     // pivot_token_idx_list (B,2) int32
    float*       out = (float*)d_out;           // (B,H) fp32

    dim3 grid(BB * 6);   // 6 blocks per batch, 8 waves (h-tiles) per block
    dim3 block(256);
    pivot_pool_wmma_kernel<<<grid, block, 0, stream>>>(X, piv, out);
}